// MultiHeadAttentionQuantum_65481071398436
// MI455X (gfx1250) — compile-verified
//
#include <hip/hip_runtime.h>
#include <hip/hip_bf16.h>
#include <math.h>

typedef __attribute__((ext_vector_type(2))) float v2f;
typedef __attribute__((ext_vector_type(8))) float v8f;

#define S_LEN 2048
#define EMB   64
#define NW    8     // wires = d_k
#define QB    128   // query rows per workgroup (8 waves x 16)

// D = A(16x4) * B(4x16) + C, all f32.  A/B = 2 VGPRs/lane, C/D = 8 VGPRs/lane.
__device__ __forceinline__ v8f wmma4(v2f a, v2f b, v8f c) {
  return __builtin_amdgcn_wmma_f32_16x16x4_f32(false, a, false, b, (short)0, c,
                                               false, false);
}

// half-wave swap (lanes i <-> i^16) in one v_permlanex16_b32
__device__ __forceinline__ float xor16(float v) {
  int i = __float_as_int(v);
  int o = __builtin_amdgcn_permlanex16(i, i, 0x76543210, 0xfedcba98, false, false);
  return __int_as_float(o);
}

// Load a 16x8 (rows x d) tile fragment pair from LDS z[row][d] for either the
// A (M=row) or B (N=row) operand of the 16x16x4 f32 WMMA.
//   frag0: K = 0..3   frag1: K = 4..7
// VGPR0: lanes 0-15 -> K=0, lanes 16-31 -> K=2 ; VGPR1: K=1 / K=3.
__device__ __forceinline__ void load_frag_pair(const float* zsm, int rowbase,
                                               int r, int hi, v2f& f0, v2f& f1) {
  const float* p = zsm + (rowbase + r) * NW + hi * 2;
  f0 = *(const v2f*)p;        // d = {hi*2, hi*2+1}
  f1 = *(const v2f*)(p + 4);  // d = {4+hi*2, 4+hi*2+1}
}

// V B-fragment raw values for one key tile: vv[2*sl], vv[2*sl+1] are the two
// VGPRs of K-slice sl.  Column clamped to r&7: lanes r>=8 produce duplicate
// (harmless, never stored) output columns -> no masking, no exec divergence.
__device__ __forceinline__ void load_vtile(const float* zsm, int kt, int rc,
                                           int hi, float vv[8]) {
#pragma unroll
  for (int sl = 0; sl < 4; ++sl) {
    int row = kt + sl * 4 + hi * 2;
    vv[2 * sl]     = zsm[row * NW + rc];
    vv[2 * sl + 1] = zsm[(row + 1) * NW + rc];
  }
}

// ---------------------------------------------------------------------------
// Kernel 1: fused quantum-head + flash attention for one (b, h, 128-query blk)
// ---------------------------------------------------------------------------
__global__ __launch_bounds__(256) void qattn_kernel(
    const float* __restrict__ x,      // [B, S, E]
    const float* __restrict__ theta,  // [8]
    float* __restrict__ attn_out) {   // [B, S, E] (pre-projection), f32
  __shared__ float zsm[S_LEN * NW];   // 64 KB: full K/V (=Q) for this (b,h)

  const int qblk = blockIdx.x & 15;   // 16 query blocks of 128
  const int bh   = blockIdx.x >> 4;
  const int b    = bh >> 3;
  const int h    = bh & 7;
  const int tid  = threadIdx.x;

  float th[8];
#pragma unroll
  for (int j = 0; j < 8; ++j) th[j] = theta[j];

  // ---- stage z = quantum_heads(x, theta) for ALL 2048 positions of (b,h) ----
  const float* xbase = x + (size_t)b * S_LEN * EMB + h * NW;
#pragma unroll
  for (int k = 0; k < 8; ++k) {
    int pos = tid + k * 256;
    const float* xrow = xbase + (size_t)pos * EMB;
    if (k + 1 < 8)
      __builtin_prefetch(xbase + (size_t)(pos + 256) * EMB, 0, 3);  // global_prefetch_b8
    float4 xa = ((const float4*)xrow)[0];
    float4 xc = ((const float4*)xrow)[1];
    float c0 = __cosf(xa.x + th[0]);
    float c1 = __cosf(xa.y + th[1]);
    float c2 = __cosf(xa.z + th[2]);
    float c3 = __cosf(xa.w + th[3]);
    float c4 = __cosf(xc.x + th[4]);
    float c5 = __cosf(xc.y + th[5]);
    float c6 = __cosf(xc.z + th[6]);
    float c7 = __cosf(xc.w + th[7]);
    float z1 = c0 * c1, z2 = z1 * c2, z3 = z2 * c3, z4 = z3 * c4;
    float z5 = z4 * c5, z6 = z5 * c6, z7 = z6 * c7;
    float z0 = ((c1 * c2) * (c3 * c4)) * ((c5 * c6) * c7);  // <Z_0>
    float* zp = zsm + pos * NW;
    ((float4*)zp)[0] = make_float4(z0, z1, z2, z3);
    ((float4*)zp)[1] = make_float4(z4, z5, z6, z7);
  }
  __syncthreads();

  const int wid  = tid >> 5;
  const int lane = tid & 31;
  const int r    = lane & 15;
  const int rc   = lane & 7;
  const int hi   = lane >> 4;
  const int qrow = qblk * QB + wid * 16;  // 16 queries per wave

  // Q^T B-fragments (K-dim = d), fixed across the whole key loop
  v2f bq0, bq1;
  load_frag_pair(zsm, qrow, r, hi, bq0, bq1);

  // scores bounded: |q.k|/sqrt(8) <= sqrt(8)  ->  constant-shift softmax,
  // exp2 argument in [-8.2, 0] so the raw v_exp_f32 is exact enough.
  constexpr float L2E = 1.4426950408889634f;
  constexpr float K1  = 0.35355339059327373f * L2E;   // (1/sqrt8)*log2(e)
  constexpr float K0  = -2.8284271247461903f * L2E;   // -sqrt8*log2(e)

  v8f   acc  = {};   // O tile (16 queries x 16 cols, d<8 used), C/D layout
  float lsum = 0.f;  // per-lane softmax denominator partial (lane = query)

  // ---- software pipeline: prefetch tile 0 ----
  v2f   ak0, ak1;
  float vv[8];
  load_frag_pair(zsm, 0, r, hi, ak0, ak1);
  load_vtile(zsm, 0, rc, hi, vv);

  for (int kt = 0; kt < S_LEN; kt += 16) {
    v2f   ck0 = ak0, ck1 = ak1;
    float cv[8];
#pragma unroll
    for (int i = 0; i < 8; ++i) cv[i] = vv[i];

    // issue next tile's LDS loads (wraps harmlessly on the last iteration)
    int nkt = (kt + 16) & (S_LEN - 1);
    load_frag_pair(zsm, nkt, r, hi, ak0, ak1);
    load_vtile(zsm, nkt, rc, hi, vv);

    // S^T tile = K * Q^T : D[t][m] = sum_d k[t][d] q[m][d]
    v8f st = {};
    st = wmma4(ck0, bq0, st);
    st = wmma4(ck1, bq1, st);

    // p = exp(score - sqrt8); lane holds query (lane&15), VGPR i holds key
    float pv[8];
#pragma unroll
    for (int i = 0; i < 8; ++i)
      pv[i] = __builtin_amdgcn_exp2f(st[i] * K1 + K0);
    lsum += ((pv[0] + pv[1]) + (pv[2] + pv[3])) +
            ((pv[4] + pv[5]) + (pv[6] + pv[7]));

    // Build P A-fragments (M=query, K=key) from the P^T register layout:
    // only a half-wave permlane swap is needed, no LDS transpose.
    float sx[8];
#pragma unroll
    for (int i = 0; i < 8; ++i) sx[i] = xor16(pv[i]);
    v2f pa[4];
    pa[0] = v2f{hi ? sx[2] : pv[0], hi ? sx[3] : pv[1]};  // t = 0..3
    pa[1] = v2f{hi ? sx[6] : pv[4], hi ? sx[7] : pv[5]};  // t = 4..7
    pa[2] = v2f{hi ? pv[2] : sx[0], hi ? pv[3] : sx[1]};  // t = 8..11
    pa[3] = v2f{hi ? pv[6] : sx[4], hi ? pv[7] : sx[5]};  // t = 12..15

    // O += P * V  (B = V tile, K = key slice; cols 8..15 are dup'd junk)
#pragma unroll
    for (int sl = 0; sl < 4; ++sl) {
      v2f bv = v2f{cv[2 * sl], cv[2 * sl + 1]};
      acc = wmma4(pa[sl], bv, acc);
    }
  }

  // finalize softmax normalization: merge half-wave partial sums
  float ltot = lsum + xor16(lsum);  // lane n: query (n&15)
  float inv  = 1.0f / ltot;

  float* outp = attn_out + (size_t)b * S_LEN * EMB + h * NW;
#pragma unroll
  for (int i = 0; i < 8; ++i) {
    float f = __shfl(inv, i + hi * 8, 32);  // broadcast 1/l of row (i + hi*8)
    float v = acc[i] * f;
    if (r < 8) outp[(size_t)(qrow + i + hi * 8) * EMB + r] = v;
  }
}

// ---------------------------------------------------------------------------
// Kernel 2: out[BS, 64] = attn_out[BS, 64] @ W^T  (W row-major [E][E])
// one wave -> 16 rows x 64 cols, 64 chained f32 WMMAs
// ---------------------------------------------------------------------------
__global__ __launch_bounds__(256) void combine_kernel(
    const float* __restrict__ a,   // [B*S, 64]
    const float* __restrict__ W,   // [64, 64]
    float* __restrict__ out) {     // [B*S, 64]
  __shared__ float Wl[EMB * EMB];  // 16 KB
  const int tid = threadIdx.x;
#pragma unroll
  for (int k = 0; k < 16; ++k) Wl[tid + k * 256] = W[tid + k * 256];
  __syncthreads();

  const int wid  = tid >> 5;
  const int lane = tid & 31;
  const int r    = lane & 15;
  const int hi   = lane >> 4;
  const int mb   = (blockIdx.x * 8 + wid) * 16;  // 512 row tiles total

  // A fragments for 16 K-slices (K = 0..63)
  v2f af[16];
#pragma unroll
  for (int i = 0; i < 16; ++i)
    af[i] = *(const v2f*)(a + (size_t)(mb + r) * EMB + 4 * i + hi * 2);

#pragma unroll
  for (int nt = 0; nt < 4; ++nt) {
    v8f accw = {};
#pragma unroll
    for (int i = 0; i < 16; ++i) {
      // B[k][n] = W[n][k] : row = output col, from LDS
      v2f bf = *(const v2f*)(Wl + (nt * 16 + r) * EMB + 4 * i + hi * 2);
      accw = wmma4(af[i], bf, accw);
    }
#pragma unroll
    for (int i = 0; i < 8; ++i)
      out[(size_t)(mb + i + hi * 8) * EMB + nt * 16 + r] = accw[i];
  }
}

extern "C" void kernel_launch(void* const* d_in, const int* in_sizes, int n_in,
                              void* d_out, int out_size, void* d_ws,
                              size_t ws_size, hipStream_t stream) {
  (void)in_sizes; (void)n_in; (void)out_size; (void)ws_size;
  const float* x     = (const float*)d_in[0];
  const float* theta = (const float*)d_in[1];
  const float* W     = (const float*)d_in[2];
  float* out  = (float*)d_out;
  float* attn = (float*)d_ws;  // [4*2048*64] f32 = 2 MiB scratch

  // 32 (b,h) problems x 16 query blocks of 128 rows
  qattn_kernel<<<dim3(512), dim3(256), 0, stream>>>(x, theta, attn);
  // 512 row tiles / 8 waves per block
  combine_kernel<<<dim3(64), dim3(256), 0, stream>>>(attn, W, out);
}